// FastRCNNLoss_89670327206586
// MI455X (gfx1250) — compile-verified
//
#include <hip/hip_runtime.h>
#include <float.h>
#include <stdint.h>

#define C_CLASSES 91

__device__ __forceinline__ float wave_reduce_sum(float v) {
#pragma unroll
    for (int off = 16; off > 0; off >>= 1) v += __shfl_xor(v, off, 32);
    return v;
}

__device__ __forceinline__ float wave_reduce_max(float v) {
#pragma unroll
    for (int off = 16; off > 0; off >>= 1) v = fmaxf(v, __shfl_xor(v, off, 32));
    return v;
}

// One wave (32 lanes) per RoI row. Grid-stride over rows.
// Emits per-block partials: [ce_sum, huber_sum, mask_count] into d_ws.
__global__ __launch_bounds__(256) void frcnn_loss_partial(
    const float* __restrict__ cls_out,   // [N, 91]
    const float* __restrict__ box_out,   // [N, 91*4]
    const int*   __restrict__ cls_tgt,   // [N]
    const float* __restrict__ box_tgt,   // [N, 4]
    float* __restrict__ partials,        // [gridDim.x * 3]
    int nRows)
{
    __shared__ float s_part[3][8];
    __shared__ float s_bt[8][4];   // async-staged box targets, one slot per wave

    const int lane          = threadIdx.x & 31;
    const int wv            = threadIdx.x >> 5;
    const int wavesPerBlock = blockDim.x >> 5;
    const int gw            = blockIdx.x * wavesPerBlock + wv;
    const int stride        = gridDim.x * wavesPerBlock;

    float acc_ce = 0.f, acc_hub = 0.f, acc_pos = 0.f;

    for (int row = gw; row < nRows; row += stride) {
        const float* cls = cls_out + (size_t)row * C_CLASSES;

        // --- Fire the 16B box-target fetch for this row into LDS early
        // (CDNA5 async global->LDS path, overlapped with the softmax work).
        if (lane == 0) {
            const float* btp = box_tgt + (size_t)row * 4;
            // Generic LDS pointers carry the LDS byte offset in the low 32 bits.
            unsigned lds_addr = (unsigned)(uintptr_t)(&s_bt[wv][0]);
            asm volatile("global_load_async_to_lds_b128 %0, %1, off"
                         :: "v"(lds_addr), "v"(btp) : "memory");
        }

        // Prefetch the next class row this wave will process.
        if (row + stride < nRows)
            __builtin_prefetch(cls_out + (size_t)(row + stride) * C_CLASSES, 0, 3);

        // --- log-softmax CE: coalesced read of the 91-logit row
        float v0 = cls[lane];
        float v1 = cls[lane + 32];
        float v2 = (lane + 64 < C_CLASSES) ? cls[lane + 64] : -FLT_MAX;

        float m = wave_reduce_max(fmaxf(fmaxf(v0, v1), v2));
        float e = __expf(v0 - m) + __expf(v1 - m)
                + ((lane + 64 < C_CLASSES) ? __expf(v2 - m) : 0.f);
        e = wave_reduce_sum(e);

        const int t = cls_tgt[row];
        // Gather x[t] from registers via cross-lane shuffles (no extra memory traffic).
        float x0 = __shfl(v0, t & 31, 32);
        float x1 = __shfl(v1, t & 31, 32);
        float x2 = __shfl(v2, t & 31, 32);
        float xt = (t >= 64) ? x2 : ((t >= 32) ? x1 : x0);
        if (lane == 0) acc_ce += __logf(e) + m - xt;

        // --- box Huber on the 4 gathered coords (positive RoIs only)
        if (lane < 4) {
            float bo = box_out[(size_t)row * (4 * C_CLASSES) + 4 * t + lane];
            asm volatile("s_wait_asynccnt 0" ::: "memory");
            float bt = s_bt[wv][lane];
            float d  = bt - bo;
            float ad = fabsf(d);
            float h  = (ad <= 1.0f) ? 0.5f * d * d : (ad - 0.5f);
            if (t > 0) {
                acc_hub += h;
                if (lane == 0) acc_pos += 4.0f;  // mask has 4 entries per positive row
            }
        }
    }

    // Per-wave then per-block deterministic reduction.
    acc_ce  = wave_reduce_sum(acc_ce);
    acc_hub = wave_reduce_sum(acc_hub);
    acc_pos = wave_reduce_sum(acc_pos);
    if (lane == 0) {
        s_part[0][wv] = acc_ce;
        s_part[1][wv] = acc_hub;
        s_part[2][wv] = acc_pos;
    }
    __syncthreads();
    if (threadIdx.x < 3) {
        float s = 0.f;
        for (int i = 0; i < wavesPerBlock; ++i) s += s_part[threadIdx.x][i];
        partials[blockIdx.x * 3 + threadIdx.x] = s;
    }
}

__global__ __launch_bounds__(256) void frcnn_loss_finalize(
    const float* __restrict__ partials, int nBlocks, float nLabels,
    float* __restrict__ out)
{
    __shared__ float s[3][8];
    float ce = 0.f, hub = 0.f, pos = 0.f;
    for (int i = threadIdx.x; i < nBlocks; i += blockDim.x) {
        ce  += partials[3 * i + 0];
        hub += partials[3 * i + 1];
        pos += partials[3 * i + 2];
    }
    ce  = wave_reduce_sum(ce);
    hub = wave_reduce_sum(hub);
    pos = wave_reduce_sum(pos);
    const int lane = threadIdx.x & 31;
    const int wv   = threadIdx.x >> 5;
    if (lane == 0) { s[0][wv] = ce; s[1][wv] = hub; s[2][wv] = pos; }
    __syncthreads();
    if (threadIdx.x == 0) {
        float tce = 0.f, thub = 0.f, tpos = 0.f;
        const int nw = blockDim.x >> 5;
        for (int i = 0; i < nw; ++i) { tce += s[0][i]; thub += s[1][i]; tpos += s[2][i]; }
        // divide_no_nan semantics
        out[0] = (nLabels != 0.f) ? tce / nLabels : 0.f;   // class_loss
        out[1] = (tpos   != 0.f) ? thub / tpos   : 0.f;   // box_loss
    }
}

extern "C" void kernel_launch(void* const* d_in, const int* in_sizes, int n_in,
                              void* d_out, int out_size, void* d_ws, size_t ws_size,
                              hipStream_t stream) {
    const float* cls_out = (const float*)d_in[0];  // [B,R,91]  fp32
    const float* box_out = (const float*)d_in[1];  // [B,R,364] fp32
    const int*   cls_tgt = (const int*)d_in[2];    // [B,R]     int32
    const float* box_tgt = (const float*)d_in[3];  // [B,R,4]   fp32

    const int nRows = in_sizes[2];                 // B*R = 65536

    int blocks = 1024;                             // 8192 waves @ 8 waves/block
    const int maxBlocks = (int)(ws_size / (3 * sizeof(float)));
    if (blocks > maxBlocks) blocks = maxBlocks > 0 ? maxBlocks : 1;
    const int wavesNeeded = (nRows + 0) ;          // keep launch even if nRows small
    (void)wavesNeeded;

    float* partials = (float*)d_ws;

    frcnn_loss_partial<<<blocks, 256, 0, stream>>>(
        cls_out, box_out, cls_tgt, box_tgt, partials, nRows);
    frcnn_loss_finalize<<<1, 256, 0, stream>>>(
        partials, blocks, (float)nRows, (float*)d_out);
}